// CustomLoraLayerMerged_33277406609704
// MI455X (gfx1250) — compile-verified
//
#include <hip/hip_runtime.h>

typedef float v2f __attribute__((ext_vector_type(2)));
typedef float v8f __attribute__((ext_vector_type(8)));

// Problem constants (from the reference)
#define BJ   8192   // B*J flattened rows
#define MDIM 4096   // d_model
#define NADP 4      // adapters
#define DRK  16     // lora rank
#define KOUT 4096   // out dim

__device__ __forceinline__ v8f wmma_f32(v2f a, v2f b, v8f c) {
  // 8 args: (neg_a, A, neg_b, B, c_mod, C, reuse_a, reuse_b)
  return __builtin_amdgcn_wmma_f32_16x16x4_f32(false, a, false, b, (short)0, c,
                                               false, false);
}

// ---------------------------------------------------------------------------
// Stage 1: t[n][row][d] = sum_m x[row][m] * lora_A[n][d][m]
// One wave per 16-row tile; all 4 adapters accumulated simultaneously so x is
// read exactly once from HBM. lora_A (1 MB) stays L2-resident.
// grid = 128 blocks x 128 threads (4 waves/block) -> 512 waves = 512 row tiles
// ---------------------------------------------------------------------------
__global__ __launch_bounds__(128) void lora_down_kernel(
    const float* __restrict__ x, const float* __restrict__ lora_A,
    float* __restrict__ t) {
  const int lane = threadIdx.x & 31;
  const int wave = threadIdx.x >> 5;
  const int rb   = blockIdx.x * 4 + wave;   // 16-row tile index (0..511)
  const int h    = lane >> 4;               // K-half select
  const int r    = lane & 15;               // A row / B col within tile

  const float* xrow = x + (size_t)(rb * 16 + r) * MDIM + 2 * h;
  const float* arow = lora_A + (size_t)r * MDIM + 2 * h;  // adapter 0, row d=r

  v8f acc0 = {}, acc1 = {}, acc2 = {}, acc3 = {};

#pragma unroll 4
  for (int c = 0; c < MDIM; c += 4) {
    v2f a  = *(const v2f*)(xrow + c);
    v2f b0 = *(const v2f*)(arow + c);
    v2f b1 = *(const v2f*)(arow + (size_t)1 * DRK * MDIM + c);
    v2f b2 = *(const v2f*)(arow + (size_t)2 * DRK * MDIM + c);
    v2f b3 = *(const v2f*)(arow + (size_t)3 * DRK * MDIM + c);
    acc0 = wmma_f32(a, b0, acc0);
    acc1 = wmma_f32(a, b1, acc1);
    acc2 = wmma_f32(a, b2, acc2);
    acc3 = wmma_f32(a, b3, acc3);
  }

  // C/D layout: VGPR vr, lane -> row (vr + 8h) of tile, col d = lane&15
  const size_t rowbase = (size_t)rb * 16 + 8 * h;
#pragma unroll
  for (int vr = 0; vr < 8; ++vr) {
    size_t row = rowbase + vr;
    t[((size_t)0 * BJ + row) * DRK + r] = acc0[vr];
    t[((size_t)1 * BJ + row) * DRK + r] = acc1[vr];
    t[((size_t)2 * BJ + row) * DRK + r] = acc2[vr];
    t[((size_t)3 * BJ + row) * DRK + r] = acc3[vr];
  }
}

// ---------------------------------------------------------------------------
// Stage 2: out[n][row][k] = sum_d t[n][row][d] * lora_B[n][k][d]
// One wave per (n, 16-row tile, 64-col tile). t tile (1 KB) loaded once as
// 4 A-frags; D=16 contraction = 4 x K4 WMMAs per 16x16 out subtile.
// Output streamed with non-temporal stores (537 MB, the bandwidth bottleneck)
// so t / lora_B stay hot in L2.
// grid = 16384 blocks x 256 threads (8 waves) -> 131072 waves = 4*512*64 tiles
// ---------------------------------------------------------------------------
__global__ __launch_bounds__(256) void lora_up_kernel(
    const float* __restrict__ t, const float* __restrict__ lora_B,
    float* __restrict__ out) {
  const int lane = threadIdx.x & 31;
  const int wid  = blockIdx.x * 8 + (threadIdx.x >> 5);
  const int cb   = wid & 63;          // 64-col tile (0..63)
  const int rb   = (wid >> 6) & 511;  // 16-row tile (0..511)
  const int n    = wid >> 15;         // adapter (0..3)
  const int h    = lane >> 4;
  const int r    = lane & 15;

  // A-frags: t[n][rb*16 + r][d], K(d) = 4c + 2h + {0,1}
  const float* trow = t + ((size_t)n * BJ + rb * 16 + r) * DRK + 2 * h;
  v2f a0 = *(const v2f*)(trow + 0);
  v2f a1 = *(const v2f*)(trow + 4);
  v2f a2 = *(const v2f*)(trow + 8);
  v2f a3 = *(const v2f*)(trow + 12);

#pragma unroll
  for (int s = 0; s < 4; ++s) {
    const int col = cb * 64 + s * 16 + r;  // B col N = lane&15
    const float* brow = lora_B + ((size_t)n * KOUT + col) * DRK + 2 * h;
    v2f b0 = *(const v2f*)(brow + 0);
    v2f b1 = *(const v2f*)(brow + 4);
    v2f b2 = *(const v2f*)(brow + 8);
    v2f b3 = *(const v2f*)(brow + 12);

    v8f acc = {};
    acc = wmma_f32(a0, b0, acc);
    acc = wmma_f32(a1, b1, acc);
    acc = wmma_f32(a2, b2, acc);
    acc = wmma_f32(a3, b3, acc);

    // out[n][rb*16 + vr + 8h][cb*64 + s*16 + (lane&15)]
    float* orow = out + ((size_t)n * BJ + rb * 16 + 8 * h) * KOUT +
                  (size_t)cb * 64 + s * 16 + r;
#pragma unroll
    for (int vr = 0; vr < 8; ++vr) {
      __builtin_nontemporal_store(acc[vr], orow + (size_t)vr * KOUT);
    }
  }
}

extern "C" void kernel_launch(void* const* d_in, const int* in_sizes, int n_in,
                              void* d_out, int out_size, void* d_ws,
                              size_t ws_size, hipStream_t stream) {
  const float* x      = (const float*)d_in[0];  // [4, 2048, 4096]
  const float* lora_A = (const float*)d_in[1];  // [4, 16, 4096]
  const float* lora_B = (const float*)d_in[2];  // [4, 4096, 16]
  float* out = (float*)d_out;                   // [4, 4, 2048, 4096]
  float* t   = (float*)d_ws;                    // [4, 8192, 16] = 2 MB scratch

  lora_down_kernel<<<128, 128, 0, stream>>>(x, lora_A, t);
  lora_up_kernel<<<16384, 256, 0, stream>>>(t, lora_B, out);
}